// LightningGNN_18159121728201
// MI455X (gfx1250) — compile-verified
//
#include <hip/hip_runtime.h>

typedef __attribute__((ext_vector_type(2))) float v2f;
typedef __attribute__((ext_vector_type(8))) float v8f;

#define F 128   // feature width of both GCN layers (F_IN == H == 128)

// ---------------------------------------------------------------- utilities
__global__ void zero_f32_kernel(float* __restrict__ p, long n) {
  long i = (long)blockIdx.x * blockDim.x + threadIdx.x;
  if (i < n) p[i] = 0.0f;
}

__global__ void deg_init_kernel(float* __restrict__ deg, int n) {
  int i = blockIdx.x * blockDim.x + threadIdx.x;
  if (i < n) deg[i] = 1.0f;            // self-loop contributes 1 to in-degree
}

__global__ void deg_count_kernel(const int* __restrict__ dst,
                                 float* __restrict__ deg, int E) {
  int e = blockIdx.x * blockDim.x + threadIdx.x;
  if (e < E) atomicAdd(&deg[dst[e]], 1.0f);
}

__global__ void deg_rsqrt_kernel(float* __restrict__ deg, int n) {
  int i = blockIdx.x * blockDim.x + threadIdx.x;
  if (i < n) {
    float d = deg[i];
    deg[i] = d > 0.0f ? rsqrtf(d) : 0.0f;   // becomes dinv in place
  }
}

// ---------------------------------------------------------- WMMA fp32 GEMM
// H = X @ W      X:[N,128]  W:[128,128] row-major  H:[N,128]
// Block = 256 threads = 8 waves. Block owns 16 rows; wave w owns cols 16w..16w+15.
// Full fp32 precision via V_WMMA_F32_16X16X4_F32 (K=4 per instruction, 32 steps).
__global__ __launch_bounds__(256) void gemm_wmma_kernel(
    const float* __restrict__ X, const float* __restrict__ W,
    float* __restrict__ H, int N) {
  __shared__ float sX[16 * F];               // 8 KB staged A tile
  const int tid  = threadIdx.x;
  const int row0 = blockIdx.x * 16;

  // Stage 16 rows of X via coalesced float4 loads (clamped for tail blocks).
  {
    const int    maxq = N * (F / 4) - 1;     // last valid float4 index in X
    const float4* Xq  = reinterpret_cast<const float4*>(X);
    float4*       sXq = reinterpret_cast<float4*>(sX);
    int q0 = row0 * (F / 4) + tid;
    int q1 = q0 + 256;
    sXq[tid]       = Xq[q0 < maxq ? q0 : maxq];
    sXq[tid + 256] = Xq[q1 < maxq ? q1 : maxq];
  }
  __syncthreads();

  const int lane = tid & 31;
  const int wave = tid >> 5;
  const int col0 = wave * 16;
  const int m    = lane & 15;   // row (A) / col (B,D) within tile
  const int h    = lane >> 4;   // lane-half: selects K pair {0,1} vs {2,3}

  v8f acc = {};
  #pragma unroll 8
  for (int k = 0; k < F; k += 4) {
    v2f a, b;
    // A 16x4 fp32 layout: lanes 0-15 -> VGPR0:K=0 VGPR1:K=1 ; lanes 16-31 -> K=2,3
    a[0] = sX[m * F + k + 2 * h + 0];
    a[1] = sX[m * F + k + 2 * h + 1];
    // B 4x16 fp32 layout mirrors: VGPR r holds row K = 2*h + r, N = lane&15
    b[0] = W[(k + 2 * h + 0) * F + col0 + m];
    b[1] = W[(k + 2 * h + 1) * F + col0 + m];
    acc = __builtin_amdgcn_wmma_f32_16x16x4_f32(
        /*neg_a=*/false, a, /*neg_b=*/false, b,
        /*c_mod=*/(short)0, acc, /*reuse_a=*/false, /*reuse_b=*/false);
  }

  // D 16x16 fp32 layout: VGPR r -> lanes 0-15: M=r, lanes 16-31: M=8+r; N = lane&15
  float* outp = H + (size_t)(row0 + h * 8) * F + col0 + m;
  if (row0 + 16 <= N) {
    // wave-uniform fast path: 8 unguarded stores with immediate offsets
    #pragma unroll
    for (int r = 0; r < 8; ++r) outp[(size_t)r * F] = acc[r];
  } else {
    #pragma unroll
    for (int r = 0; r < 8; ++r)
      if (row0 + h * 8 + r < N) outp[(size_t)r * F] = acc[r];
  }
}

// ---------------------------------------------------- edge scatter (message passing)
// One wave per edge (self-loops appended as eid in [E, E+N)). Lane covers 4 floats:
// coalesced float4 gather of h[src], scaled by norm, 4 fp32 global atomics to agg[dst].
// Edge index is wave-uniform: force scalarization so index/norm fetches ride the
// scalar path and VMEM is reserved for the feature gather + atomics.
__global__ void scatter_kernel(const float* __restrict__ hx,
                               const int* __restrict__ src,
                               const int* __restrict__ dst,
                               const float* __restrict__ dinv,
                               float* __restrict__ agg, int E, int N) {
  long gid = (long)blockIdx.x * blockDim.x + threadIdx.x;
  int eid  = __builtin_amdgcn_readfirstlane((int)(gid >> 5));  // wave-uniform
  int lane = (int)(gid & 31);
  if (eid >= E + N) return;                                    // uniform branch
  int s, d;
  if (eid < E) { s = src[eid]; d = dst[eid]; }
  else         { s = eid - E;  d = s; }                        // self-loop
  float nrm = dinv[s] * dinv[d];
  const float4* hs = reinterpret_cast<const float4*>(hx + (size_t)s * F);
  float4 v = hs[lane];
  float* out = agg + (size_t)d * F + lane * 4;
  atomicAdd(out + 0, v.x * nrm);
  atomicAdd(out + 1, v.y * nrm);
  atomicAdd(out + 2, v.z * nrm);
  atomicAdd(out + 3, v.w * nrm);
}

__global__ void bias_relu_kernel(float* __restrict__ hm,
                                 const float* __restrict__ b, long total) {
  long i = (long)blockIdx.x * blockDim.x + threadIdx.x;
  if (i < total) {
    float v = hm[i] + b[(int)(i & (F - 1))];
    hm[i] = v > 0.0f ? v : 0.0f;
  }
}

// -------------------------------------------------------------- mean pool
__global__ void pool_kernel(const float* __restrict__ hm,
                            const int* __restrict__ batch,
                            float* __restrict__ sums, float* __restrict__ cnt,
                            int N) {
  long gid = (long)blockIdx.x * blockDim.x + threadIdx.x;
  int node = __builtin_amdgcn_readfirstlane((int)(gid >> 5)); // wave-uniform
  int lane = (int)(gid & 31);
  if (node >= N) return;                                      // uniform branch
  int g = batch[node];
  const float4* hv = reinterpret_cast<const float4*>(hm + (size_t)node * F);
  float4 v = hv[lane];
  float* out = sums + (size_t)g * F + lane * 4;
  atomicAdd(out + 0, v.x);
  atomicAdd(out + 1, v.y);
  atomicAdd(out + 2, v.z);
  atomicAdd(out + 3, v.w);
  if (lane == 0) atomicAdd(&cnt[g], 1.0f);
}

// -------------------------------------------------------------- head: [G,128]@[128,2]+bl
__global__ void head_kernel(const float* __restrict__ sums,
                            const float* __restrict__ cnt,
                            const float* __restrict__ Wl,
                            const float* __restrict__ bl,
                            float* __restrict__ out, int G) {
  int t = blockIdx.x * blockDim.x + threadIdx.x;
  if (t >= G * 2) return;
  int g = t >> 1, o = t & 1;
  float c = cnt[g];
  c = c > 1.0f ? c : 1.0f;
  float inv = 1.0f / c;
  float acc = 0.0f;
  #pragma unroll 8
  for (int f = 0; f < F; ++f) acc += sums[g * F + f] * Wl[f * 2 + o];
  out[t] = acc * inv + bl[o];
}

// ---------------------------------------------------------------- launcher
extern "C" void kernel_launch(void* const* d_in, const int* in_sizes, int n_in,
                              void* d_out, int out_size, void* d_ws, size_t ws_size,
                              hipStream_t stream) {
  const float* x     = (const float*)d_in[0];
  const int*   ei    = (const int*)  d_in[1];   // [2,E] row-major: src then dst
  const int*   batch = (const int*)  d_in[2];
  const float* W1    = (const float*)d_in[3];
  const float* b1    = (const float*)d_in[4];
  const float* W2    = (const float*)d_in[5];
  const float* b2    = (const float*)d_in[6];
  const float* Wl    = (const float*)d_in[7];
  const float* bl    = (const float*)d_in[8];
  float*       out   = (float*)d_out;

  const int N = in_sizes[0] / F;
  const int E = in_sizes[1] / 2;
  const int G = out_size / 2;
  const int* src = ei;
  const int* dst = ei + E;

  // workspace layout (floats, 16B-aligned sections)
  float* ws = (float*)d_ws;
  size_t off = 0;
  float* dinv = ws + off; off += (size_t)((N + 3) & ~3);
  float* bufA = ws + off; off += (size_t)N * F;     // GEMM outputs (hx, hx2)
  float* bufB = ws + off; off += (size_t)N * F;     // aggregation / activations
  float* sums = ws + off; off += (size_t)G * F;
  float* cnt  = ws + off; off += (size_t)((G + 3) & ~3);
  (void)ws_size; (void)n_in;

  const int T = 256;
  const long NF = (long)N * F;
  auto nb = [](long n, int t) { return (int)((n + t - 1) / t); };
  const int gemm_blocks    = (N + 15) / 16;
  const long scatter_thr   = (long)(E + N) * 32;
  const long pool_thr      = (long)N * 32;

  // --- degree / symmetric norm ---
  deg_init_kernel <<<nb(N, T), T, 0, stream>>>(dinv, N);
  deg_count_kernel<<<nb(E, T), T, 0, stream>>>(dst, dinv, E);
  deg_rsqrt_kernel<<<nb(N, T), T, 0, stream>>>(dinv, N);

  // --- layer 1: h = relu(scatter(norm * (x@W1)[src] -> dst) + b1) ---
  zero_f32_kernel <<<nb(NF, T), T, 0, stream>>>(bufB, NF);
  gemm_wmma_kernel<<<gemm_blocks, T, 0, stream>>>(x, W1, bufA, N);
  scatter_kernel  <<<nb(scatter_thr, T), T, 0, stream>>>(bufA, src, dst, dinv, bufB, E, N);
  bias_relu_kernel<<<nb(NF, T), T, 0, stream>>>(bufB, b1, NF);

  // --- layer 2 ---
  gemm_wmma_kernel<<<gemm_blocks, T, 0, stream>>>(bufB, W2, bufA, N);
  zero_f32_kernel <<<nb(NF, T), T, 0, stream>>>(bufB, NF);
  scatter_kernel  <<<nb(scatter_thr, T), T, 0, stream>>>(bufA, src, dst, dinv, bufB, E, N);
  bias_relu_kernel<<<nb(NF, T), T, 0, stream>>>(bufB, b2, NF);

  // --- global mean pool + linear head ---
  zero_f32_kernel <<<nb((long)G * F + G, T), T, 0, stream>>>(sums, (long)G * F + G); // sums then cnt (contiguous)
  pool_kernel     <<<nb(pool_thr, T), T, 0, stream>>>(bufB, batch, sums, cnt, N);
  head_kernel     <<<nb(G * 2, 128), 128, 0, stream>>>(sums, cnt, Wl, bl, out, G);
}